// GraphCNN_9045201125817
// MI455X (gfx1250) — compile-verified
//
#include <hip/hip_runtime.h>
#include <hip/hip_bf16.h>

typedef __attribute__((ext_vector_type(2))) float v2f;
typedef __attribute__((ext_vector_type(8))) float v8f;

#define NNODES 50000
#define HDIM   128
#define NLAYERS 3
#define NROWTILES (NNODES / 16)   // 3125

// ---------------- utility kernels ----------------

__global__ void k_zero(float* __restrict__ p, int n) {
  int i = blockIdx.x * blockDim.x + threadIdx.x;
  if (i < n) p[i] = 0.0f;
}

__global__ void k_degcount(const int* __restrict__ dst, float* __restrict__ deg, int ne) {
  int e = blockIdx.x * blockDim.x + threadIdx.x;
  if (e < ne) atomicAdd(&deg[dst[e]], 1.0f);
}

__global__ void k_invert(float* __restrict__ deg, int n) {
  int i = blockIdx.x * blockDim.x + threadIdx.x;
  if (i < n) deg[i] = 1.0f / deg[i];
}

// One wave per edge: 32 lanes x float4 = one 512B row. Gather h[src], atomic-add
// into pooled[dst]. Both matrices are 25.6MB -> L2 resident on the 192MB L2.
__global__ void k_spmm(const float* __restrict__ h, const int* __restrict__ src,
                       const int* __restrict__ dst, float* __restrict__ pooled, int ne) {
  int lane = threadIdx.x & 31;
  int e = blockIdx.x * (blockDim.x >> 5) + (threadIdx.x >> 5);
  if (e >= ne) return;
  int s = src[e], d = dst[e];
  const float4* hv = (const float4*)(h + (size_t)s * HDIM);
  float4 v = hv[lane];
  float* p = pooled + (size_t)d * HDIM + lane * 4;
  atomicAdd(p + 0, v.x);
  atomicAdd(p + 1, v.y);
  atomicAdd(p + 2, v.z);
  atomicAdd(p + 3, v.w);
}

// Column sum / sum-of-squares partial reduction for BatchNorm statistics.
// blockDim = 128 (one column per thread), rows chunked per block, 128-address atomics.
__global__ void k_colstats(const float* __restrict__ Y, float* __restrict__ stats,
                           int nrows, int rowsPerBlock) {
  int c = threadIdx.x;
  int r0 = blockIdx.x * rowsPerBlock;
  int r1 = r0 + rowsPerBlock; if (r1 > nrows) r1 = nrows;
  float s = 0.0f, s2 = 0.0f;
  for (int r = r0; r < r1; ++r) {
    float v = Y[(size_t)r * HDIM + c];
    s += v;
    s2 = fmaf(v, v, s2);
  }
  atomicAdd(&stats[c], s);
  atomicAdd(&stats[HDIM + c], s2);
}

// Fold BN into per-channel affine: BN(x) = a*x + b.  one block of 128 threads.
__global__ void k_bnfinal(const float* __restrict__ stats, const float* __restrict__ gamma,
                          const float* __restrict__ beta, float* __restrict__ ab, float invN) {
  int c = threadIdx.x;
  float mean = stats[c] * invN;
  float var  = stats[HDIM + c] * invN - mean * mean;
  float inv  = rsqrtf(var + 1e-5f);
  float a = gamma[c] * inv;
  ab[c] = a;
  ab[HDIM + c] = fmaf(-mean, a, beta[c]);
}

// out = relu(a*Y + b)  elementwise (outer BN + ReLU).
__global__ void k_apply(const float* __restrict__ Y, const float* __restrict__ ab,
                        float* __restrict__ out, int total) {
  int i = blockIdx.x * blockDim.x + threadIdx.x;
  if (i >= total) return;
  int c = i & (HDIM - 1);
  out[i] = fmaxf(fmaf(Y[i], ab[c], ab[HDIM + c]), 0.0f);
}

// ---------------- WMMA fp32 GEMM ----------------
// Y[nrows x 128] = preprocess(A[nrows x 128]) @ W[128 x 128] + bias
//   MODE 0 : preprocess = row * invdeg[row]                (GEMM-1, folds avg pooling)
//   MODE 1 : preprocess = relu(a[k]*x + b[k])              (GEMM-2, folds BN1+ReLU)
// One wave owns a 16-column tile of W (64 VGPRs, reused across row tiles) and
// grid-strides over 16-row tiles. 32 x V_WMMA_F32_16X16X4_F32 per tile (K=128).
// A-frag layout (ISA 7.12.2, 32-bit A 16x4): lane<16 -> K=0,1 ; lane>=16 -> K=2,3
// of row M=lane&15. B mirrored with N across lanes. C/D: M = r + 8*(lane>>4).
template <int MODE>
__global__ void k_gemm(const float* __restrict__ A, const float* __restrict__ W,
                       const float* __restrict__ bias, const float* __restrict__ rowscale,
                       const float* __restrict__ ab, float* __restrict__ Y, int nrowtiles) {
  const int lane  = threadIdx.x & 31;
  const int gwave = blockIdx.x * (blockDim.x >> 5) + (threadIdx.x >> 5);
  const int nwav  = gridDim.x * (blockDim.x >> 5);
  const int half  = lane >> 4;      // 0: K/M low half, 1: high half
  const int lid   = lane & 15;
  const int ncol  = blockIdx.y * 16 + lid;

  v2f bfrag[32];
#pragma unroll
  for (int ks = 0; ks < 32; ++ks) {
    int k = ks * 4 + half * 2;
    bfrag[ks].x = W[(size_t)k * HDIM + ncol];
    bfrag[ks].y = W[(size_t)(k + 1) * HDIM + ncol];
  }
  float bb = bias[ncol];

  for (int rt = gwave; rt < nrowtiles; rt += nwav) {   // wave-uniform: EXEC all-ones
    int arow = rt * 16 + lid;
    const float* Arow = A + (size_t)arow * HDIM;
    float rs = (MODE == 0) ? rowscale[arow] : 0.0f;
    v8f c = {};
#pragma unroll
    for (int ks = 0; ks < 32; ++ks) {
      int k = ks * 4 + half * 2;
      float x0 = Arow[k];
      float x1 = Arow[k + 1];
      if (MODE == 0) {
        x0 *= rs; x1 *= rs;
      } else {
        x0 = fmaxf(fmaf(x0, ab[k],     ab[HDIM + k]),     0.0f);
        x1 = fmaxf(fmaf(x1, ab[k + 1], ab[HDIM + k + 1]), 0.0f);
      }
      v2f a; a.x = x0; a.y = x1;
      c = __builtin_amdgcn_wmma_f32_16x16x4_f32(false, a, false, bfrag[ks],
                                                (short)0, c, false, false);
    }
#pragma unroll
    for (int r = 0; r < 8; ++r) {
      int orow = rt * 16 + half * 8 + r;
      Y[(size_t)orow * HDIM + ncol] = c[r] + bb;
    }
  }
}

// ---------------- driver ----------------

extern "C" void kernel_launch(void* const* d_in, const int* in_sizes, int n_in,
                              void* d_out, int out_size, void* d_ws, size_t ws_size,
                              hipStream_t stream) {
  const float* feats = (const float*)d_in[0];
  const float* W1    = (const float*)d_in[1];   // L x 128 x 128
  const float* b1    = (const float*)d_in[2];   // L x 128
  const float* g1    = (const float*)d_in[3];
  const float* be1   = (const float*)d_in[4];
  const float* W2    = (const float*)d_in[5];
  const float* b2    = (const float*)d_in[6];
  const float* og    = (const float*)d_in[7];
  const float* ob    = (const float*)d_in[8];
  const int*   src   = (const int*)d_in[9];
  const int*   dst   = (const int*)d_in[10];
  const int ne = in_sizes[9];                   // E + N self loops = 850000

  // workspace layout (~51.5 MB)
  float* bufA   = (float*)d_ws;                 // pooled / Y2
  float* bufB   = bufA + (size_t)NNODES * HDIM; // Y1 / h
  float* invdeg = bufB + (size_t)NNODES * HDIM; // NNODES
  float* statsA = invdeg + NNODES;              // 256 (sum | sumsq)
  float* statsB = statsA + 2 * HDIM;            // 256
  float* abA    = statsB + 2 * HDIM;            // 256 (a | b)
  float* abB    = abA + 2 * HDIM;               // 256

  const int total = NNODES * HDIM;
  const float invN = 1.0f / (float)NNODES;
  const int rowsPerBlock = 500;
  const int statBlocks = (NNODES + rowsPerBlock - 1) / rowsPerBlock;
  dim3 gemmGrid(98, HDIM / 16, 1);              // 784 waves per column tile

  // degree (constant across layers)
  k_zero<<<(NNODES + 255) / 256, 256, 0, stream>>>(invdeg, NNODES);
  k_degcount<<<(ne + 255) / 256, 256, 0, stream>>>(dst, invdeg, ne);
  k_invert<<<(NNODES + 255) / 256, 256, 0, stream>>>(invdeg, NNODES);

  for (int l = 0; l < NLAYERS; ++l) {
    const float* hin = (l == 0) ? feats : bufB;
    // clear accumulation targets (pooled + both stats regions, contiguous 512 floats)
    k_zero<<<(total + 255) / 256, 256, 0, stream>>>(bufA, total);
    k_zero<<<2, 256, 0, stream>>>(statsA, 4 * HDIM);

    // pooled = Adj @ h   (divide-by-degree folded into GEMM-1)
    k_spmm<<<(ne + 7) / 8, 256, 0, stream>>>(hin, src, dst, bufA, ne);

    // Y1 = (pooled * invdeg) @ W1[l] + b1[l]
    k_gemm<0><<<gemmGrid, 256, 0, stream>>>(bufA, W1 + (size_t)l * HDIM * HDIM,
                                            b1 + (size_t)l * HDIM, invdeg, nullptr,
                                            bufB, NROWTILES);
    // BN1 statistics -> affine
    k_colstats<<<statBlocks, HDIM, 0, stream>>>(bufB, statsA, NNODES, rowsPerBlock);
    k_bnfinal<<<1, HDIM, 0, stream>>>(statsA, g1 + (size_t)l * HDIM,
                                      be1 + (size_t)l * HDIM, abA, invN);

    // Y2 = relu(BN1(Y1)) @ W2[l] + b2[l]   (BN1+ReLU folded into A-operand load)
    k_gemm<1><<<gemmGrid, 256, 0, stream>>>(bufB, W2 + (size_t)l * HDIM * HDIM,
                                            b2 + (size_t)l * HDIM, nullptr, abA,
                                            bufA, NROWTILES);
    // outer BN statistics -> affine
    k_colstats<<<statBlocks, HDIM, 0, stream>>>(bufA, statsB, NNODES, rowsPerBlock);
    k_bnfinal<<<1, HDIM, 0, stream>>>(statsB, og + (size_t)l * HDIM,
                                      ob + (size_t)l * HDIM, abB, invN);

    // h = relu(BN(Y2))
    float* hout = (l == NLAYERS - 1) ? (float*)d_out : bufB;
    k_apply<<<(total + 255) / 256, 256, 0, stream>>>(bufA, abB, hout, total);
  }
}